// Regressor_LabelAttention_65403761984354
// MI455X (gfx1250) — compile-verified
//
#include <hip/hip_runtime.h>
#include <hip/hip_bf16.h>

typedef __attribute__((ext_vector_type(2))) float v2f;
typedef __attribute__((ext_vector_type(8))) float v8f;

#define NB 64
#define NC 2205
#define ND 512

// Builtin expects int* pointees (diag: "__device__ int *"), i.e. addrspace(1)/(3) int*.
#define ASG(p) ((__attribute__((address_space(1))) int*)(p))
#define ASL(p) ((__attribute__((address_space(3))) int*)(p))

__global__ __launch_bounds__(256) void label_attn_kernel(
    const float* __restrict__ feature, const float* __restrict__ attribute,
    const float* __restrict__ Wq, const float* __restrict__ bq,
    const float* __restrict__ Wk, const float* __restrict__ bk,
    const float* __restrict__ Wv, const float* __restrict__ bv,
    const float* __restrict__ Wp, float* __restrict__ out)
{
    __shared__ float lds_feat[2208];   // feature row for this b
    __shared__ float lds_G[256];       // 16x16 Gram matrix of parameter vectors
    __shared__ float redm[8];          // per-wave max
    __shared__ float reds[16];         // per-wave [sum_e | sum_fe]

    const int tid  = threadIdx.x;
    const int b    = blockIdx.x;
    const int lane = tid & 31;
    const int wid  = tid >> 5;

    if (wid == 0) {
        // ---- Wave 0: Gram matrix G = M^T M via V_WMMA_F32_16X16X4_F32 ----
        // M is 512 x 16, columns: 0=Wq 1=bq 2=Wk 3=bk 4=Wv 5=bv 6=Wp, 7..15 = 0
        const int m  = lane & 15;
        const int kb = (lane >> 4) << 1;   // 0 for lanes 0-15, 2 for lanes 16-31
        const float* col;
        switch (m) {
            case 0: col = Wq; break;
            case 1: col = bq; break;
            case 2: col = Wk; break;
            case 3: col = bk; break;
            case 4: col = Wv; break;
            case 5: col = bv; break;
            case 6: col = Wp; break;
            default: col = Wq; break;      // masked to zero below
        }
        const float msk = (m < 7) ? 1.0f : 0.0f;
#if __has_builtin(__builtin_amdgcn_wmma_f32_16x16x4_f32)
        v8f acc = {0.f, 0.f, 0.f, 0.f, 0.f, 0.f, 0.f, 0.f};
        for (int d0 = 0; d0 < ND; d0 += 4) {
            v2f ab;
            ab.x = msk * col[d0 + kb];
            ab.y = msk * col[d0 + kb + 1];
            // A[m][k] = M[d0+k][m], B[k][n] = M[d0+k][n]; identical lane contents.
            acc = __builtin_amdgcn_wmma_f32_16x16x4_f32(
                false, ab, false, ab, (short)0, acc, false, false);
        }
        // D layout: VGPR r, lanes 0-15 -> row r, lanes 16-31 -> row r+8; col = lane&15
        {
            const int row0 = (lane >> 4) << 3;
            const int nn   = lane & 15;
            #pragma unroll
            for (int r = 0; r < 8; ++r) lds_G[(row0 + r) * 16 + nn] = acc[r];
        }
#else
        // Scalar fallback: compute only the Gram entries we need.
        float p0 = 0.f, p1 = 0.f, p2 = 0.f, p3 = 0.f, p4 = 0.f;
        for (int d = lane; d < ND; d += 32) {
            p0 += Wq[d] * Wk[d];
            p1 += Wq[d] * bk[d] + Wk[d] * bq[d];
            p2 += bq[d] * bk[d];
            p3 += Wv[d] * Wp[d];
            p4 += bv[d] * Wp[d];
        }
        #pragma unroll
        for (int off = 16; off > 0; off >>= 1) {
            p0 += __shfl_xor(p0, off, 32);
            p1 += __shfl_xor(p1, off, 32);
            p2 += __shfl_xor(p2, off, 32);
            p3 += __shfl_xor(p3, off, 32);
            p4 += __shfl_xor(p4, off, 32);
        }
        if (lane == 0) {
            lds_G[0 * 16 + 2] = p0;  lds_G[0 * 16 + 3] = p1;
            lds_G[2 * 16 + 1] = 0.f; lds_G[1 * 16 + 3] = p2;
            lds_G[4 * 16 + 6] = p3;  lds_G[5 * 16 + 6] = p4;
        }
        (void)msk; (void)kb; (void)col;
#endif
    } else {
        // ---- Waves 1..7: async DMA the feature row into LDS (CDNA5 path) ----
        const float* g = feature + (size_t)b * NC;
        const int t = tid - 32;                 // 0..223
#if __has_builtin(__builtin_amdgcn_global_load_async_to_lds_b32) && \
    __has_builtin(__builtin_amdgcn_s_wait_asynccnt)
        for (int c = t; c < NC; c += 224) {
            __builtin_amdgcn_global_load_async_to_lds_b32(
                ASG(const_cast<float*>(g) + c), ASL(lds_feat + c), 0, 0);
        }
#else
        for (int c = t; c < NC; c += 224) lds_feat[c] = g[c];
#endif
    }
#if __has_builtin(__builtin_amdgcn_s_wait_asynccnt)
    __builtin_amdgcn_s_wait_asynccnt(0);
#endif
    __syncthreads();

    // ---- Per-b scalars from Gram entries ----
    const float alpha = lds_G[0 * 16 + 2];                       // Wq.Wk
    const float beta  = lds_G[0 * 16 + 3] + lds_G[2 * 16 + 1];   // Wq.bk + Wk.bq
    const float gmma  = lds_G[1 * 16 + 3];                       // bq.bk
    const float delta = lds_G[4 * 16 + 6];                       // Wv.Wp
    const float eps   = lds_G[5 * 16 + 6];                       // bv.Wp
    const float a = attribute[b];
    const float s = fmaf(fmaf(alpha, a, beta), a, gmma);  // alpha*a^2 + beta*a + gamma (TEMPER=1)
    const float u = fmaf(delta, a, eps);

    // ---- Softmax statistics over the row (in LDS) ----
    float lmax = -3.4028235e38f;
    for (int c = tid; c < NC; c += 256) lmax = fmaxf(lmax, lds_feat[c] * s);
    #pragma unroll
    for (int off = 16; off > 0; off >>= 1) lmax = fmaxf(lmax, __shfl_xor(lmax, off, 32));
    if (lane == 0) redm[wid] = lmax;
    __syncthreads();
    float mx = redm[0];
    #pragma unroll
    for (int w = 1; w < 8; ++w) mx = fmaxf(mx, redm[w]);

    float se = 0.f, sf = 0.f;
    for (int c = tid; c < NC; c += 256) {
        const float f = lds_feat[c];
        const float e = __expf(fmaf(f, s, -mx));
        se += e;
        sf += e * f;
    }
    #pragma unroll
    for (int off = 16; off > 0; off >>= 1) {
        se += __shfl_xor(se, off, 32);
        sf += __shfl_xor(sf, off, 32);
    }
    if (lane == 0) { reds[wid] = se; reds[8 + wid] = sf; }
    __syncthreads();
    float SE = 0.f, SF = 0.f;
    #pragma unroll
    for (int w = 0; w < 8; ++w) { SE += reds[w]; SF += reds[8 + w]; }

    const float add = (SF / SE) * u;   // t_b * u_b

    // ---- out[b,c,0] = feature[b,c] + t_b*u_b ----
    float* o = out + (size_t)b * NC;
    for (int c = tid; c < NC; c += 256) o[c] = lds_feat[c] + add;
}

extern "C" void kernel_launch(void* const* d_in, const int* in_sizes, int n_in,
                              void* d_out, int out_size, void* d_ws, size_t ws_size,
                              hipStream_t stream) {
    const float* feature   = (const float*)d_in[0];
    const float* attribute = (const float*)d_in[1];
    const float* Wq = (const float*)d_in[2];
    const float* bq = (const float*)d_in[3];
    const float* Wk = (const float*)d_in[4];
    const float* bk = (const float*)d_in[5];
    const float* Wv = (const float*)d_in[6];
    const float* bv = (const float*)d_in[7];
    const float* Wp = (const float*)d_in[8];
    float* out = (float*)d_out;
    hipLaunchKernelGGL(label_attn_kernel, dim3(NB), dim3(256), 0, stream,
                       feature, attribute, Wq, bq, Wk, bk, Wv, bv, Wp, out);
}